// ConvexSurfaceSampler_5222680232169
// MI455X (gfx1250) — compile-verified
//
#include <hip/hip_runtime.h>
#include <math.h>

typedef float v2f __attribute__((ext_vector_type(2)));
typedef float v8f __attribute__((ext_vector_type(8)));

#define NTOT   8066      // number of directions
#define NB     4
#define NP     32
#define NV     64
#define BP     (NB*NP)   // 128
#define N_CHUNK 128      // directions per block (8 waves x 16)
#define NBLK_N  64       // ceil(NTOT / N_CHUNK)

// float offsets inside d_out (tuple concatenated in return order)
#define POINTS_OFF  0
#define POINTS_CNT  (BP * NTOT * 3)        // 3,097,344
#define DIRS_OFF    (POINTS_OFF + POINTS_CNT)
#define DIRS_CNT    (NTOT * 3)             // 24,198
#define LV_OFF      (DIRS_OFF + DIRS_CNT)
#define LV_CNT      (BP * NV * 3)          // 24,576
#define DHDZ_OFF    (LV_OFF + LV_CNT)
#define DHDZ_CNT    ((size_t)BP * NV * NTOT)  // 66,076,672
#define ZMS_OFF     (DHDZ_OFF + DHDZ_CNT)

// hardware v_exp_f32 (2^x); __exp2f does not exist as a HIP device intrinsic
__device__ __forceinline__ float hw_exp2(float x) { return __builtin_amdgcn_exp2f(x); }

__global__ void ConvexSampler_dirs_kernel(float* __restrict__ dirs) {
    int n = blockIdx.x * blockDim.x + threadIdx.x;
    if (n >= NTOT) return;
    const float PI = 3.14159265358979323846f;
    int i1, i2;
    if (n < 8064)      { i1 = n / 128 + 1; i2 = n % 128; } // body rows th1[1:-1]
    else if (n == 8064){ i1 = 0;  i2 = 0; }                // pole0
    else               { i1 = 64; i2 = 0; }                // pole1
    float th1 = -0.5f * PI + (float)i1 * (PI / 64.0f);     // linspace(-pi/2, pi/2, 65)
    float th2 = -PI        + (float)i2 * (PI / 64.0f);     // linspace(-pi,   pi, 129)
    float s1, c1, s2, c2;
    sincosf(th1, &s1, &c1);   // precise: dirs is itself a validated output
    sincosf(th2, &s2, &c2);
    dirs[n * 3 + 0] = c1 * c2;
    dirs[n * 3 + 1] = c1 * s2;
    dirs[n * 3 + 2] = s1;
}

__global__ __launch_bounds__(256) void ConvexSampler_main_kernel(
    const float* __restrict__ vertices,   // [4,32,64,3]
    const float* __restrict__ smooth,     // [4,32]
    const float* __restrict__ dirs,       // [NTOT,3]
    float* __restrict__ points,           // [BP*NTOT*3]
    float* __restrict__ out_lv,           // [BP*NV*3]
    float* __restrict__ dhdz_out,         // [BP,NV,NTOT]
    float* __restrict__ zms_out)          // [BP,NV,NTOT]
{
    __shared__ float s_vert[NV * 3];
    __shared__ float s_lv[NV * 3];
    __shared__ float s_mean[3];

    const int bp    = blockIdx.y;
    const int chunk = blockIdx.x;
    const int tid   = threadIdx.x;

    // ---- mean_v and local_v for this (b,p) ----
    if (tid < NV * 3) s_vert[tid] = vertices[(size_t)bp * NV * 3 + tid];
    __syncthreads();
    if (tid < 3) {
        float s = 0.f;
        for (int v = 0; v < NV; ++v) s += s_vert[v * 3 + tid];
        s_mean[tid] = s * (1.0f / NV);
    }
    __syncthreads();
    if (tid < NV * 3) {
        float lv = s_vert[tid] - s_mean[tid % 3];
        s_lv[tid] = lv;
        if (chunk == 0) out_lv[(size_t)bp * NV * 3 + tid] = lv;  // local_v output
    }
    __syncthreads();

    const float p    = smooth[bp];
    const int wave   = tid >> 5;
    const int lane   = tid & 31;
    const int half   = lane >> 4;     // 0: lanes 0-15, 1: lanes 16-31
    const int nl     = lane & 15;
    const int n0     = chunk * N_CHUNK + wave * 16;
    if (n0 >= NTOT) return;           // wave-uniform: EXEC stays all-1s for WMMA
    const int  n      = n0 + nl;
    const int  nc     = (n < NTOT) ? n : (NTOT - 1);
    const bool nvalid = (n < NTOT);

    // ---- B matrix: dirs^T tile, 4x16 f32 (K x N). K=3 padded with 0. ----
    // ISA layout: lanes 0-15 hold K=0 (v0) / K=1 (v1); lanes 16-31 hold K=2 / K=3.
    v2f bmat;
    {
        float d0 = dirs[nc * 3 + 0], d1 = dirs[nc * 3 + 1], d2 = dirs[nc * 3 + 2];
        bmat[0] = half ? d2  : d0;
        bmat[1] = half ? 0.f : d1;
    }

    // ---- z = local_v @ dirs^T via 4x V_WMMA_F32_16X16X4_F32 (v tiles of 16) ----
    // D layout: VGPR r, lanes 0-15 -> M=r ; lanes 16-31 -> M=8+r ; N = lane%16.
    // => element i = t*8+r of zv[] corresponds to v = t*16 + half*8 + r.
    float zv[32];
    #pragma unroll
    for (int t = 0; t < 4; ++t) {
        int vrow = t * 16 + nl;       // A row M = lane%16
        v2f amat;
        float l0 = s_lv[vrow * 3 + 0], l1 = s_lv[vrow * 3 + 1], l2v = s_lv[vrow * 3 + 2];
        amat[0] = half ? l2v : l0;    // K=0 / K=2
        amat[1] = half ? 0.f : l1;    // K=1 / K=3 (pad)
        v8f c = {};
        c = __builtin_amdgcn_wmma_f32_16x16x4_f32(
                /*neg_a=*/false, amat, /*neg_b=*/false, bmat,
                /*c_mod=*/(short)0, c, /*reuse_a=*/false, /*reuse_b=*/false);
        #pragma unroll
        for (int r = 0; r < 8; ++r) zv[t * 8 + r] = c[r];
    }

    // ---- max_v(relu(z));  log10 is monotone so log10(max) == max(log10) ----
    float zmax = 0.f;
    #pragma unroll
    for (int i = 0; i < 32; ++i) zmax = fmaxf(zmax, fmaxf(zv[i], 0.f));
    zmax = fmaxf(zmax, __shfl_xor(zmax, 16, 32));

    // scaling factor k = 10^clip(ceil(lk), 0, 1e20)   (hardware v_log/v_exp paths)
    float expo = __log10f(zmax) * p;          // -inf if zmax == 0 (matches ref)
    float lk   = (expo < -20.f) ? (-20.f - expo) / p : 0.f;
    float kc   = fminf(fmaxf(ceilf(lk), 0.f), 1e20f);
    float kf   = __exp10f(kc);

    // ---- pass 2: zms (store), l2(zms) (keep), zm_p sum over v ----
    const size_t base_vn = (size_t)bp * NV * NTOT;
    float l2s[32];
    float ssum = 0.f;
    #pragma unroll
    for (int t = 0; t < 4; ++t) {
        #pragma unroll
        for (int r = 0; r < 8; ++r) {
            int   i   = t * 8 + r;
            int   v   = t * 16 + half * 8 + r;
            float zm  = fmaxf(zv[i], 0.f);
            float zms = zm * kf;
            float l2  = __log2f(zms);         // v_log_f32: -inf when zms == 0
            l2s[i] = l2;
            float zp = hw_exp2(p * l2);       // zms^p via v_exp_f32
            zp = fminf(fmaxf(zp, 1e-20f), 1e20f);
            if (zms == 0.f) zp = 0.f;         // clip(...) - (zms==0)*LB
            ssum += zp;
            if (nvalid) zms_out[base_vn + (size_t)v * NTOT + n] = zms;
        }
    }
    ssum += __shfl_xor(ssum, 16, 32);
    float ls = __log2f(ssum) / p;             // log2(sum_zm) = log2(S^(1/p))

    // ---- pass 3: dhdz = (zms/sum_zm)^(p-1) (store), dhdx accumulation ----
    float acc0 = 0.f, acc1 = 0.f, acc2 = 0.f;
    const float pm1 = p - 1.f;
    #pragma unroll
    for (int t = 0; t < 4; ++t) {
        #pragma unroll
        for (int r = 0; r < 8; ++r) {
            int   i  = t * 8 + r;
            int   v  = t * 16 + half * 8 + r;
            float dh = hw_exp2(pm1 * (l2s[i] - ls)); // exp2(-inf)=0 handles zms==0
            if (nvalid) dhdz_out[base_vn + (size_t)v * NTOT + n] = dh;
            acc0 += dh * s_lv[v * 3 + 0];
            acc1 += dh * s_lv[v * 3 + 1];
            acc2 += dh * s_lv[v * 3 + 2];
        }
    }
    acc0 += __shfl_xor(acc0, 16, 32);
    acc1 += __shfl_xor(acc1, 16, 32);
    acc2 += __shfl_xor(acc2, 16, 32);

    if (half == 0 && nvalid) {
        size_t pidx = ((size_t)bp * NTOT + n) * 3;   // points[b, p*NTOT+n, :]
        points[pidx + 0] = acc0 + s_mean[0];
        points[pidx + 1] = acc1 + s_mean[1];
        points[pidx + 2] = acc2 + s_mean[2];
    }
}

extern "C" void kernel_launch(void* const* d_in, const int* in_sizes, int n_in,
                              void* d_out, int out_size, void* d_ws, size_t ws_size,
                              hipStream_t stream) {
    const float* vertices = (const float*)d_in[0];   // (4,32,64,3) f32
    const float* smooth   = (const float*)d_in[1];   // (4,32) f32
    float* out = (float*)d_out;

    float* dirs = out + DIRS_OFF;
    ConvexSampler_dirs_kernel<<<(NTOT + 255) / 256, 256, 0, stream>>>(dirs);

    dim3 grid(NBLK_N, BP);
    ConvexSampler_main_kernel<<<grid, 256, 0, stream>>>(
        vertices, smooth, dirs,
        out + POINTS_OFF, out + LV_OFF, out + DHDZ_OFF, out + ZMS_OFF);
}